// LocationAttention_3796751089824
// MI455X (gfx1250) — compile-verified
//
#include <hip/hip_runtime.h>
#include <hip/hip_bf16.h>

typedef __attribute__((ext_vector_type(2))) float v2f;
typedef __attribute__((ext_vector_type(8))) float v8f;

#define EPS 1e-9f

// ---------------------------------------------------------------------------
// Pass 1: a[tok] = exp(dot(x[tok,:], W) + bias) via V_WMMA_F32_16X16X4_F32.
// One wave32 handles 16 tokens; C/D is 16x16 where every column n holds the
// per-token (row) dot product. blockDim=256 -> 8 waves -> 128 tokens/block.
// ---------------------------------------------------------------------------
__global__ __launch_bounds__(256)
void la_pass1_scores(const float* __restrict__ x, const float* __restrict__ W,
                     const float* __restrict__ bias, float* __restrict__ a,
                     int H) {
    const int lane = threadIdx.x & 31;
    const int wave = threadIdx.x >> 5;
    const long tokBase = (long)blockIdx.x * 128 + (long)wave * 16;

    const int row = lane & 15;          // token within the 16-token tile
    const int kq  = (lane >> 4) * 2;    // lanes 0-15 cover K=0,1; lanes 16-31 K=2,3
    const float* rowPtr = x + (tokBase + row) * (long)H + kq;

    v8f acc = {0.f, 0.f, 0.f, 0.f, 0.f, 0.f, 0.f, 0.f};
    for (int k = 0; k < H; k += 4) {
        v2f aF = *(const v2f*)(rowPtr + k);            // x[row][k+kq], x[row][k+kq+1]
        v2f bF;
        bF.x = W[k + kq];                               // broadcast into all 16 columns
        bF.y = W[k + kq + 1];
        acc = __builtin_amdgcn_wmma_f32_16x16x4_f32(
            /*neg_a=*/false, aF, /*neg_b=*/false, bF,
            /*c_mod=*/(short)0, acc, /*reuse_a=*/false, /*reuse_b=*/false);
    }

    // C/D layout: lane 0 column holds tokens 0..7 (VGPR r -> M=r),
    // lane 16 column holds tokens 8..15.
    const float bb = bias[0];
    if (lane == 0 || lane == 16) {
        const long base = tokBase + (lane ? 8 : 0);
        #pragma unroll
        for (int i = 0; i < 8; ++i)
            a[base + i] = expf(acc[i] + bb);
    }
}

// ---------------------------------------------------------------------------
// Pass 2: per-batch exclusive scan of a over S=4096; store 1/(excl_sum + eps)
// so the hot output pass multiplies instead of divides.
// One 1024-thread block per batch; each thread owns 4 consecutive elements.
// ---------------------------------------------------------------------------
__global__ __launch_bounds__(1024)
void la_pass2_scan(const float* __restrict__ a, float* __restrict__ dinv, int S) {
    __shared__ float sums[1024];
    const int b = blockIdx.x;
    const int t = threadIdx.x;
    const float* ab = a + (long)b * S;

    float v0 = ab[t * 4 + 0], v1 = ab[t * 4 + 1];
    float v2 = ab[t * 4 + 2], v3 = ab[t * 4 + 3];
    sums[t] = v0 + v1 + v2 + v3;
    __syncthreads();

    // Hillis-Steele inclusive scan over 1024 thread partials.
    for (int off = 1; off < 1024; off <<= 1) {
        float add = (t >= off) ? sums[t - off] : 0.0f;
        __syncthreads();
        sums[t] += add;
        __syncthreads();
    }

    float excl = (t == 0) ? 0.0f : sums[t - 1];
    float* ob = dinv + (long)b * S;
    ob[t * 4 + 0] = 1.0f / (excl + EPS);
    ob[t * 4 + 1] = 1.0f / (excl + v0 + EPS);
    ob[t * 4 + 2] = 1.0f / (excl + v0 + v1 + EPS);
    ob[t * 4 + 3] = 1.0f / (excl + v0 + v1 + v2 + EPS);
}

// ---------------------------------------------------------------------------
// Pass 3a: chunk partial sums  Ppart[b,c,h] = sum_{s in chunk c} a[b,s]*x[b,s,h]
// grid = B*C*(H/256); fully coalesced streaming read of x.
// ---------------------------------------------------------------------------
__global__ __launch_bounds__(256)
void la_pass3a_partials(const float* __restrict__ x, const float* __restrict__ a,
                        float* __restrict__ Ppart, int S, int H, int C, int CL) {
    const int HT = H / 256;
    const int ht = blockIdx.x % HT;
    const int c  = (blockIdx.x / HT) % C;
    const int b  = blockIdx.x / (HT * C);
    const int h  = ht * 256 + threadIdx.x;

    const long sBase = (long)c * CL;
    const float* xp = x + ((long)b * S + sBase) * H + h;
    const float* ap = a + (long)b * S + sBase;

    float acc = 0.0f;
    for (int s = 0; s < CL; ++s) {
        __builtin_prefetch(xp + (long)(s + 8) * H, 0, 0);   // global_prefetch_b8
        acc = fmaf(ap[s], xp[(long)s * H], acc);
    }
    Ppart[((long)b * C + c) * H + h] = acc;
}

// ---------------------------------------------------------------------------
// Pass 3b: exclusive scan over the C chunk partials per (b,h).
// grid = B*(H/256); loads across threads are coalesced per chunk row.
// ---------------------------------------------------------------------------
__global__ __launch_bounds__(256)
void la_pass3b_scanchunks(const float* __restrict__ Ppart, float* __restrict__ Pexc,
                          int H, int C) {
    const int HT = H / 256;
    const int ht = blockIdx.x % HT;
    const int b  = blockIdx.x / HT;
    const int h  = ht * 256 + threadIdx.x;

    const float* pp = Ppart + (long)b * C * H + h;
    float* pe       = Pexc  + (long)b * C * H + h;
    float run = 0.0f;
    for (int c = 0; c < C; ++c) {
        pe[(long)c * H] = run;
        run += pp[(long)c * H];
    }
}

// ---------------------------------------------------------------------------
// Pass 3c: final streaming pass.
// out[b,s,h] = (Pexc[b,c,h] + sum_{j in chunk, j<s} a_j*x_jh) * dinv[b,s]
// ---------------------------------------------------------------------------
__global__ __launch_bounds__(256)
void la_pass3c_final(const float* __restrict__ x, const float* __restrict__ a,
                     const float* __restrict__ dinv, const float* __restrict__ Pexc,
                     float* __restrict__ out, int S, int H, int C, int CL) {
    const int HT = H / 256;
    const int ht = blockIdx.x % HT;
    const int c  = (blockIdx.x / HT) % C;
    const int b  = blockIdx.x / (HT * C);
    const int h  = ht * 256 + threadIdx.x;

    const long sBase = (long)c * CL;
    const float* xp = x   + ((long)b * S + sBase) * H + h;
    float* op       = out + ((long)b * S + sBase) * H + h;
    const float* ap = a    + (long)b * S + sBase;
    const float* dp = dinv + (long)b * S + sBase;

    float acc = Pexc[((long)b * C + c) * H + h];
    for (int s = 0; s < CL; ++s) {
        __builtin_prefetch(xp + (long)(s + 8) * H, 0, 0);   // global_prefetch_b8
        float xv = xp[(long)s * H];
        op[(long)s * H] = acc * dp[s];      // exclusive: before adding current term
        acc = fmaf(ap[s], xv, acc);
    }
}

// ---------------------------------------------------------------------------
extern "C" void kernel_launch(void* const* d_in, const int* in_sizes, int n_in,
                              void* d_out, int out_size, void* d_ws, size_t ws_size,
                              hipStream_t stream) {
    const float* x    = (const float*)d_in[0];   // [B,S,H]
    const float* W    = (const float*)d_in[1];   // [H,1]
    const float* bias = (const float*)d_in[2];   // [1]
    float* out = (float*)d_out;

    const int B = 16, S = 4096, H = 768;
    const int C = 32;            // chunks along S
    const int CL = S / C;        // 128 tokens per chunk
    const int HT = H / 256;      // 3 h-tiles of 256

    float* ws   = (float*)d_ws;
    float* a    = ws;                               // B*S
    float* dinv = a    + (size_t)B * S;             // B*S
    float* Pp   = dinv + (size_t)B * S;             // B*C*H
    float* Pe   = Pp   + (size_t)B * C * H;         // B*C*H  (total ~3.5 MB)

    // 1) scores via WMMA: 128 tokens per 256-thread block
    la_pass1_scores<<<(B * S) / 128, 256, 0, stream>>>(x, W, bias, a, H);

    // 2) per-batch scan of a -> reciprocal denominators
    la_pass2_scan<<<B, 1024, 0, stream>>>(a, dinv, S);

    // 3) chunked scan of a*x and final output
    la_pass3a_partials<<<B * C * HT, 256, 0, stream>>>(x, a, Pp, S, H, C, CL);
    la_pass3b_scanchunks<<<B * HT, 256, 0, stream>>>(Pp, Pe, H, C);
    la_pass3c_final<<<B * C * HT, 256, 0, stream>>>(x, a, dinv, Pe, out, S, H, C, CL);
}